// MjCambrianNonDifferentiableOptics_45999099740477
// MI455X (gfx1250) — compile-verified
//
#include <hip/hip_runtime.h>
#include <math.h>

// ---------------- problem constants ----------------
#define NN   257                // PSF grid (2*128+1, prime)
#define PP   272                // padded to 17*16 for WMMA tiling
#define PP2  (PP * PP)
#define IMGW 128
#define PSF_OFF (IMGW * IMGW * 3)   // out layout: img (49152) then psf (198147)
#define PI_D 3.14159265358979323846

typedef float v2f __attribute__((ext_vector_type(2)));
typedef float v8f __attribute__((ext_vector_type(8)));

// ---------------- z1 = mean(depth) ----------------
__global__ void k_depth_mean(const float* __restrict__ depth, int n, float* __restrict__ z1) {
    __shared__ float sm[256];
    float s = 0.f;
    for (int i = threadIdx.x; i < n; i += 256) s += depth[i];
    sm[threadIdx.x] = s;
    __syncthreads();
    for (int o = 128; o > 0; o >>= 1) {
        if (threadIdx.x < o) sm[threadIdx.x] += sm[threadIdx.x + o];
        __syncthreads();
    }
    if (threadIdx.x == 0) z1[0] = sm[0] / (float)n;
}

// ---------------- DFT matrix W[k][n] = exp(-2*pi*i*k*n/257), zero padded ----------------
__global__ void k_build_w(float* __restrict__ Wr, float* __restrict__ Wi) {
    int idx = blockIdx.x * 256 + threadIdx.x;
    if (idx >= PP2) return;
    int i = idx / PP, j = idx % PP;
    float cr = 0.f, ci = 0.f;
    if (i < NN && j < NN) {
        int t = (int)(((long long)i * (long long)j) % NN);   // exact angle reduction
        float ang = (float)(-2.0 * PI_D * (double)t / (double)NN);
        float sv, cv;
        sincosf(ang, &sv, &cv);
        cr = cv; ci = sv;
    }
    Wr[idx] = cr; Wi[idx] = ci;
}

// ---------------- X = fftshift(u1),  u1 = exp(i*k*sqrt(R2+z1^2)) * aperture ----------------
__global__ void k_field(const float* __restrict__ z1p, double lam,
                        float* __restrict__ Xr, float* __restrict__ Xi) {
    int idx = blockIdx.x * 256 + threadIdx.x;
    if (idx >= PP2) return;
    int i = idx / PP, j = idx % PP;
    float re = 0.f, im = 0.f;
    if (i < NN && j < NN) {
        int si = (i + 129) % NN;           // fftshift source row
        int sj = (j + 129) % NN;           // fftshift source col
        double x = -0.005 + (double)sj * (0.01 / 256.0);   // linspace(-Lx/2, Lx/2, 257)
        double y = -0.005 + (double)si * (0.01 / 256.0);
        double r2 = x * x + y * y;
        double apr = 0.5 * (0.01 / 257.0) * 128.0 + 1e-7;  // aperture radius + eps
        if (sqrt(r2) <= apr) {
            double z = (double)z1p[0];
            double ph = (2.0 * PI_D / lam) * sqrt(r2 + z * z);
            ph = fmod(ph, 2.0 * PI_D);                     // huge phase: reduce in f64
            double sv, cv;
            sincos(ph, &sv, &cv);
            re = (float)cv; im = (float)sv;
        }
    }
    Xr[idx] = re; Xi[idx] = im;
}

// ---------------- complex GEMM via V_WMMA_F32_16X16X4_F32 ----------------
// C = (Ar + i*sA*Ai) * (Br + i*sB*Bi) * scale      (272x272x272, zero-padded)
// Cr = scale*(Ar.Br - sA*sB*Ai.Bi) ; Ci = scale*(sB*Ar.Bi + sA*Ai.Br)
__global__ void k_cgemm(const float* __restrict__ Ar, const float* __restrict__ Ai,
                        const float* __restrict__ Br, const float* __restrict__ Bi,
                        float* __restrict__ Cr, float* __restrict__ Ci,
                        float sA, float sB, float scale) {
    int tn = blockIdx.x;                                    // N tile
    int tm = blockIdx.y * blockDim.y + threadIdx.y;         // M tile (wave-uniform guard)
    if (tm * 16 >= PP) return;
    int lane = threadIdx.x;
    int m    = lane & 15;
    int sel  = lane >> 4;                                   // 0: K 0,1  1: K 2,3
    int rowA = tm * 16 + m;
    int col  = tn * 16 + m;
    const float* arp = Ar + (size_t)rowA * PP;
    const float* aip = Ai + (size_t)rowA * PP;

    v8f acc_rr = {0.f,0.f,0.f,0.f,0.f,0.f,0.f,0.f};
    v8f acc_ii = {0.f,0.f,0.f,0.f,0.f,0.f,0.f,0.f};
    v8f acc_ri = {0.f,0.f,0.f,0.f,0.f,0.f,0.f,0.f};
    v8f acc_ir = {0.f,0.f,0.f,0.f,0.f,0.f,0.f,0.f};

    for (int k0 = 0; k0 < PP; k0 += 4) {
        int ka = k0 + 2 * sel;
        v2f a_r = { arp[ka], arp[ka + 1] };                       // A 16x4: row in-lane
        v2f a_i = { aip[ka], aip[ka + 1] };
        v2f b_r = { Br[(size_t)ka * PP + col], Br[(size_t)(ka + 1) * PP + col] };  // B 4x16
        v2f b_i = { Bi[(size_t)ka * PP + col], Bi[(size_t)(ka + 1) * PP + col] };
        acc_rr = __builtin_amdgcn_wmma_f32_16x16x4_f32(false, a_r, false, b_r, (short)0, acc_rr, false, false);
        acc_ii = __builtin_amdgcn_wmma_f32_16x16x4_f32(false, a_i, false, b_i, (short)0, acc_ii, false, false);
        acc_ri = __builtin_amdgcn_wmma_f32_16x16x4_f32(false, a_r, false, b_i, (short)0, acc_ri, false, false);
        acc_ir = __builtin_amdgcn_wmma_f32_16x16x4_f32(false, a_i, false, b_r, (short)0, acc_ir, false, false);
    }

    float s = sA * sB;
#pragma unroll
    for (int v = 0; v < 8; ++v) {                           // C layout: row = v + 8*sel
        int row = tm * 16 + sel * 8 + v;
        size_t o = (size_t)row * PP + col;
        Cr[o] = scale * (acc_rr[v] - s * acc_ii[v]);
        Ci[o] = scale * (sB * acc_ri[v] + sA * acc_ir[v]);
    }
}

// ---------------- U2 = fftshift(H) .* U1  (H computed on the fly) ----------------
__global__ void k_applyH(const float* __restrict__ Ur, const float* __restrict__ Ui,
                         float* __restrict__ Or, float* __restrict__ Oi, double lam) {
    int idx = blockIdx.x * 256 + threadIdx.x;
    if (idx >= PP2) return;
    int i = idx / PP, j = idx % PP;
    float outr = 0.f, outi = 0.f;
    if (i < NN && j < NN) {
        int si = (i + 129) % NN;
        int sj = (j + 129) % NN;
        double fx = ((double)sj - 128.0) * (25700.0 / 256.0);   // linspace(-fmax,fmax,257)
        double fy = ((double)si - 128.0) * (25700.0 / 256.0);
        double il = 1.0 / lam;
        double hr = 0.0, hi = 0.0;
        if (fx * fx + fy * fy < il * il) {                      // evanescent cutoff
            double a = 1.0 - (lam * fx) * (lam * fx) - (lam * fy) * (lam * fy);
            if (a < 0.0) a = 0.0;
            double ph = (2.0 * PI_D / lam) * 0.05 * sqrt(a);    // k*z*sqrt(arg)
            ph = fmod(ph, 2.0 * PI_D);
            double sv, cv;
            sincos(ph, &sv, &cv);
            hr = cv; hi = sv;
        }
        double ur = (double)Ur[idx], ui = (double)Ui[idx];
        outr = (float)(hr * ur - hi * ui);
        outi = (float)(hr * ui + hi * ur);
    }
    Or[idx] = outr; Oi[idx] = outi;
}

// ---------------- psf_raw = |ifftshift(Y)|^2 -> out, partial sums -> sumbuf ----------------
__global__ void k_psfmag(const float* __restrict__ Yr, const float* __restrict__ Yi,
                         float* __restrict__ psf, int c, float* __restrict__ sumbuf) {
    __shared__ float sm[256];
    int idx = blockIdx.x * 256 + threadIdx.x;
    float v = 0.f;
    if (idx < NN * NN) {
        int i = idx / NN, j = idx % NN;
        int si = (i + 128) % NN;           // ifftshift
        int sj = (j + 128) % NN;
        float re = Yr[(size_t)si * PP + sj];
        float im = Yi[(size_t)si * PP + sj];
        v = re * re + im * im;
        psf[(size_t)idx * 3 + c] = v;
    }
    sm[threadIdx.x] = v;
    __syncthreads();
    for (int o = 128; o > 0; o >>= 1) {
        if (threadIdx.x < o) sm[threadIdx.x] += sm[threadIdx.x + o];
        __syncthreads();
    }
    if (threadIdx.x == 0) sumbuf[blockIdx.x] = sm[0];
}

// ---------------- psf /= (sum + 1e-7) ----------------
__global__ void k_psfnorm(float* __restrict__ psf, int c,
                          const float* __restrict__ sumbuf, int nb) {
    __shared__ float sm[256];
    __shared__ float inv;
    float s = 0.f;
    for (int i = threadIdx.x; i < nb; i += 256) s += sumbuf[i];
    sm[threadIdx.x] = s;
    __syncthreads();
    for (int o = 128; o > 0; o >>= 1) {
        if (threadIdx.x < o) sm[threadIdx.x] += sm[threadIdx.x + o];
        __syncthreads();
    }
    if (threadIdx.x == 0) inv = 1.f / (sm[0] + 1e-7f);
    __syncthreads();
    float iv = inv;
    for (int idx = threadIdx.x; idx < NN * NN; idx += 256)
        psf[(size_t)idx * 3 + c] *= iv;
}

// ---------------- depthwise SAME correlation: out[y,x,c] = sum img[iy,ix,c]*psf[iy-y+128,ix-x+128,c]
// Image row staged into LDS via CDNA5 async DMA (GLOBAL_LOAD_ASYNC_TO_LDS_B32, ASYNCcnt).
__global__ void k_conv(const float* __restrict__ img, const float* __restrict__ psf,
                       float* __restrict__ out) {
    __shared__ float simg[IMGW];
    int c = blockIdx.z;
    int x = blockIdx.x * 16 + threadIdx.x;
    int y = blockIdx.y * 16 + threadIdx.y;
    int tid = threadIdx.y * 16 + threadIdx.x;
    float acc = 0.f;
    for (int iy = 0; iy < IMGW; ++iy) {
        __syncthreads();                                    // protect previous iteration's reads
        if (tid < IMGW) {                                   // waves 0-3 fully active (EXEC all-1)
            unsigned lds_addr = (unsigned)(size_t)&simg[tid];
            const float* gptr = img + ((size_t)iy * IMGW + tid) * 3 + c;
            asm volatile("global_load_async_to_lds_b32 %0, %1, off"
                         :: "v"(lds_addr), "v"(gptr) : "memory");
        }
        asm volatile("s_wait_asynccnt 0x0" ::: "memory");   // DMA complete in issuing wave
        __syncthreads();                                    // publish LDS row to all waves
        int dy = iy - y + 128;                              // always in [1,255]
        const float* prow = psf + ((size_t)dy * NN + (128 - x)) * 3 + c;
        __builtin_prefetch(prow, 0, 1);                     // global_prefetch_b8 (L2-resident psf)
        float a = 0.f;
#pragma unroll 4
        for (int ix = 0; ix < IMGW; ++ix)
            a = fmaf(simg[ix], prow[(size_t)ix * 3], a);
        acc += a;
    }
    out[((size_t)y * IMGW + x) * 3 + c] = acc;
}

// ---------------- host launcher ----------------
extern "C" void kernel_launch(void* const* d_in, const int* in_sizes, int n_in,
                              void* d_out, int out_size, void* d_ws, size_t ws_size,
                              hipStream_t stream) {
    const float* img   = (const float*)d_in[0];   // (128,128,3) f32
    const float* depth = (const float*)d_in[1];   // (128,128)   f32
    float* out = (float*)d_out;                   // img (49152) ++ psf (198147)
    float* ws  = (float*)d_ws;

    float* z1     = ws;                 // [0]
    float* sumbuf = ws + 8;             // up to 512 partials
    float* Wr = ws + 1024;              // 8 buffers of PP*PP f32 (~2.3 MB total)
    float* Wi = Wr + PP2;
    float* Xr = Wi + PP2;
    float* Xi = Xr + PP2;
    float* Tr = Xi + PP2;
    float* Ti = Tr + PP2;
    float* Yr = Ti + PP2;
    float* Yi = Yr + PP2;
    float* psf_out = out + PSF_OFF;

    k_depth_mean<<<1, 256, 0, stream>>>(depth, in_sizes[1], z1);

    int gb = (PP2 + 255) / 256;
    k_build_w<<<gb, 256, 0, stream>>>(Wr, Wi);

    dim3 gg(PP / 16, (PP / 16 + 3) / 4);    // 17 x 5 blocks
    dim3 bb(32, 4);                         // 4 waves per block, one 16x16 tile each
    int gp = (NN * NN + 255) / 256;         // 259 blocks
    const float inv_n2 = 1.0f / ((float)NN * (float)NN);
    const double lambdas[3] = {610e-9, 530e-9, 470e-9};

    for (int c = 0; c < 3; ++c) {
        double lam = lambdas[c];
        // X = fftshift(aperture field)
        k_field<<<gb, 256, 0, stream>>>(z1, lam, Xr, Xi);
        // U1 = W * X * W           (fft2)
        k_cgemm<<<gg, bb, 0, stream>>>(Wr, Wi, Xr, Xi, Tr, Ti, 1.f, 1.f, 1.f);
        k_cgemm<<<gg, bb, 0, stream>>>(Tr, Ti, Wr, Wi, Yr, Yi, 1.f, 1.f, 1.f);
        // U2 = fftshift(H) .* U1
        k_applyH<<<gb, 256, 0, stream>>>(Yr, Yi, Xr, Xi, lam);
        // u2 = (1/N^2) conj(W) * U2 * conj(W)      (ifft2)
        k_cgemm<<<gg, bb, 0, stream>>>(Wr, Wi, Xr, Xi, Tr, Ti, -1.f, 1.f, 1.f);
        k_cgemm<<<gg, bb, 0, stream>>>(Tr, Ti, Wr, Wi, Yr, Yi, 1.f, -1.f, inv_n2);
        // psf = |ifftshift(u2)|^2 ; normalize
        k_psfmag<<<gp, 256, 0, stream>>>(Yr, Yi, psf_out, c, sumbuf);
        k_psfnorm<<<1, 256, 0, stream>>>(psf_out, c, sumbuf, gp);
    }

    // depthwise SAME cross-correlation with normalized PSF
    k_conv<<<dim3(8, 8, 3), dim3(16, 16), 0, stream>>>(img, psf_out, out);
}